// GNNEncoder_31679678775439
// MI455X (gfx1250) — compile-verified
//
#include <hip/hip_runtime.h>
#include <stdint.h>

#define N_NODES 40000
#define N_EDGES 640000
#define D 128

typedef __attribute__((ext_vector_type(2))) float v2f;
typedef __attribute__((ext_vector_type(8))) float v8f;

// ---------------------------------------------------------------------------
// Utility: grid-stride zero fill
// ---------------------------------------------------------------------------
__global__ void zero_f32(float* __restrict__ p, int n) {
  int i = blockIdx.x * blockDim.x + threadIdx.x;
  int stride = gridDim.x * blockDim.x;
  for (; i < n; i += stride) p[i] = 0.0f;
}

// ---------------------------------------------------------------------------
// Edge degree count (float so we can atomicAdd and later invert in place)
// ---------------------------------------------------------------------------
__global__ void count_edges(const long long* __restrict__ ei, float* __restrict__ cnt) {
  int e = blockIdx.x * blockDim.x + threadIdx.x;
  if (e >= N_EDGES) return;
  int dst = (int)ei[N_EDGES + e];
  atomicAdd(&cnt[dst], 1.0f);
}

__global__ void inv_counts(float* __restrict__ cnt) {
  int i = blockIdx.x * blockDim.x + threadIdx.x;
  if (i < N_NODES) cnt[i] = 1.0f / fmaxf(cnt[i], 1.0f);
}

// ---------------------------------------------------------------------------
// Scatter-add of neighbor features: one wave per edge, lane owns 4 floats.
// summed[dst, :] += feat[src, :]   (RMW atomics land in the 192MB L2)
// ---------------------------------------------------------------------------
__global__ void scatter_sum(const float* __restrict__ feat,
                            const long long* __restrict__ ei,
                            float* __restrict__ summed) {
  long long tid = (long long)blockIdx.x * blockDim.x + threadIdx.x;
  int e = (int)(tid >> 5);
  if (e >= N_EDGES) return;
  int lane = (int)(tid & 31);
  int src = (int)ei[e];
  int dst = (int)ei[N_EDGES + e];
  const float4 v = *(const float4*)(feat + (long long)src * D + lane * 4);
  float* base = summed + (long long)dst * D + lane * 4;
  atomicAdd(base + 0, v.x);
  atomicAdd(base + 1, v.y);
  atomicAdd(base + 2, v.z);
  atomicAdd(base + 3, v.w);
}

// ---------------------------------------------------------------------------
// Fused SAGEConv GEMM using fp32 WMMA (V_WMMA_F32_16X16X4_F32):
//   out[n,:] = L2norm( (summed[n,:]*invcnt[n]) @ Wl + bl + xin[n,:] @ Wr )
//   optional ReLU after normalize (layer 0).
// 4 waves per WG; each wave computes a 16-row block x all 128 cols (8 tiles).
// Wl/Wr staged into dynamic LDS (128 KB < 320 KB WGP budget) with CDNA5
// async global->LDS copies (GLOBAL_LOAD_ASYNC_TO_LDS_B128, ASYNCcnt).
//
// fp32 WMMA fragment layout (wave32):
//   A (16x4):  lane L -> row M = L%16;  VGPR0/1 = K = (L<16 ? 0,1 : 2,3)
//   B (4x16):  lane L -> col N = L%16;  VGPR0/1 = K = (L<16 ? 0,1 : 2,3)
//   C/D:       VGPR r, lane L -> row M = r + 8*(L/16), col N = L%16
// ---------------------------------------------------------------------------
__launch_bounds__(128)
__global__ void sage_gemm(const float* __restrict__ xin,
                          const float* __restrict__ summed,
                          const float* __restrict__ invcnt,
                          const float* __restrict__ Wl,
                          const float* __restrict__ bl,
                          const float* __restrict__ Wr,
                          float* __restrict__ out,
                          int do_relu) {
  extern __shared__ float lds[];
  float* sWl = lds;            // D*D floats
  float* sWr = lds + D * D;    // D*D floats

  // --- Async global->LDS staging of both weight matrices -------------------
  // LDS byte offset = low 32 bits of the flat LDS pointer (aperture layout:
  // addr[31:0] is the 0-based LDS address).
  {
    const uint32_t ldsWl = (uint32_t)(uintptr_t)sWl;
    const uint32_t ldsWr = (uint32_t)(uintptr_t)sWr;
    for (int i = threadIdx.x; i < (D * D) / 4; i += blockDim.x) {
      uint64_t gl = (uint64_t)(uintptr_t)(Wl + i * 4);
      uint64_t gr = (uint64_t)(uintptr_t)(Wr + i * 4);
      uint32_t ll = ldsWl + (uint32_t)i * 16u;
      uint32_t lr = ldsWr + (uint32_t)i * 16u;
      asm volatile("global_load_async_to_lds_b128 %0, %1, off"
                   :
                   : "v"(ll), "v"(gl)
                   : "memory");
      asm volatile("global_load_async_to_lds_b128 %0, %1, off"
                   :
                   : "v"(lr), "v"(gr)
                   : "memory");
    }
    asm volatile("s_wait_asynccnt 0x0" ::: "memory");
  }
  __syncthreads();

  const int lane  = threadIdx.x & 31;
  const int wave  = threadIdx.x >> 5;
  const int ncol  = lane & 15;            // A-row M / B-col N within tile
  const int khalf = (lane >> 4) << 1;     // 0 (lanes 0-15) or 2 (lanes 16-31)
  const long long row0 = ((long long)blockIdx.x * 4 + wave) * 16;

  const float ic      = invcnt[row0 + ncol];           // mean scale for lane's A-row
  const float* aggRow = summed + (row0 + ncol) * D;
  const float* xRow   = xin    + (row0 + ncol) * D;

  // Accumulators initialized with the bias (depends only on column N)
  v8f acc[8];
#pragma unroll
  for (int t = 0; t < 8; ++t) {
    float bv = bl[t * 16 + ncol];
#pragma unroll
    for (int r = 0; r < 8; ++r) acc[t][r] = bv;
  }

#pragma unroll 2
  for (int k0 = 0; k0 < D; k0 += 4) {
    v2f a_agg = *(const v2f*)(aggRow + k0 + khalf);
    a_agg *= ic;                                        // fold mean into A frag
    v2f a_x = *(const v2f*)(xRow + k0 + khalf);

    // Load ALL B fragments for this k-step first so the LDS reads batch up
    // behind one dscnt wait, then the 16 WMMAs issue back-to-back.
    v2f bL[8], bR[8];
#pragma unroll
    for (int t = 0; t < 8; ++t) {
      const int bc = t * 16 + ncol;
      bL[t].x = sWl[(k0 + khalf) * D + bc];
      bL[t].y = sWl[(k0 + khalf + 1) * D + bc];
      bR[t].x = sWr[(k0 + khalf) * D + bc];
      bR[t].y = sWr[(k0 + khalf + 1) * D + bc];
    }
    // Two independent t-sweeps: consecutive WMMAs hit different accumulators,
    // so same-C RAW pairs are 8 instructions apart.
#pragma unroll
    for (int t = 0; t < 8; ++t)
      acc[t] = __builtin_amdgcn_wmma_f32_16x16x4_f32(
          false, a_agg, false, bL[t], (short)0, acc[t], false, false);
#pragma unroll
    for (int t = 0; t < 8; ++t)
      acc[t] = __builtin_amdgcn_wmma_f32_16x16x4_f32(
          false, a_x, false, bR[t], (short)0, acc[t], false, false);
  }

  // Row-wise L2 normalize (+ optional ReLU) and store.
  // Row for accumulator slot r: M = r + 8*(lane/16); columns live in lane%16
  // across the 8 tiles, so the norm reduction is a 16-lane butterfly.
#pragma unroll
  for (int r = 0; r < 8; ++r) {
    float ss = 0.0f;
#pragma unroll
    for (int t = 0; t < 8; ++t) ss += acc[t][r] * acc[t][r];
    ss += __shfl_xor(ss, 1, 32);
    ss += __shfl_xor(ss, 2, 32);
    ss += __shfl_xor(ss, 4, 32);
    ss += __shfl_xor(ss, 8, 32);
    float inv = 1.0f / fmaxf(sqrtf(ss), 1e-12f);
    long long row = row0 + r + ((lane >> 4) << 3);
    float* orow = out + row * D;
#pragma unroll
    for (int t = 0; t < 8; ++t) {
      float v = acc[t][r] * inv;
      if (do_relu) v = fmaxf(v, 0.0f);
      orow[t * 16 + ncol] = v;
    }
  }
}

// ---------------------------------------------------------------------------
// Launch: counts once, then per layer: zero + scatter + fused WMMA GEMM.
// ---------------------------------------------------------------------------
extern "C" void kernel_launch(void* const* d_in, const int* in_sizes, int n_in,
                              void* d_out, int out_size, void* d_ws, size_t ws_size,
                              hipStream_t stream) {
  const float*     x   = (const float*)d_in[0];
  const long long* ei  = (const long long*)d_in[1];
  const float*     Wl0 = (const float*)d_in[2];
  const float*     bl0 = (const float*)d_in[3];
  const float*     Wr0 = (const float*)d_in[4];
  const float*     Wl1 = (const float*)d_in[5];
  const float*     bl1 = (const float*)d_in[6];
  const float*     Wr1 = (const float*)d_in[7];
  float*           out = (float*)d_out;

  float* cnt    = (float*)d_ws;                       // 40000 (padded to 40960)
  float* summed = cnt + 40960;                        // N_NODES * D
  float* h      = summed + (size_t)N_NODES * D;       // N_NODES * D

  const int ldsBytes = 2 * D * D * (int)sizeof(float);  // 128 KB

  // Degrees -> inverse counts (shared by both layers)
  zero_f32<<<256, 256, 0, stream>>>(cnt, N_NODES);
  count_edges<<<(N_EDGES + 255) / 256, 256, 0, stream>>>(ei, cnt);
  inv_counts<<<(N_NODES + 255) / 256, 256, 0, stream>>>(cnt);

  // ---- Layer 0: h = relu(normalize(mean_agg(x) @ Wl0 + bl0 + x @ Wr0)) ----
  zero_f32<<<2048, 256, 0, stream>>>(summed, N_NODES * D);
  scatter_sum<<<(N_EDGES * 32) / 256, 256, 0, stream>>>(x, ei, summed);
  sage_gemm<<<N_NODES / 64, 128, ldsBytes, stream>>>(x, summed, cnt, Wl0, bl0, Wr0, h, 1);

  // ---- Layer 1: out = normalize(mean_agg(h) @ Wl1 + bl1 + h @ Wr1) --------
  zero_f32<<<2048, 256, 0, stream>>>(summed, N_NODES * D);
  scatter_sum<<<(N_EDGES * 32) / 256, 256, 0, stream>>>(h, ei, summed);
  sage_gemm<<<N_NODES / 64, 128, ldsBytes, stream>>>(h, summed, cnt, Wl1, bl1, Wr1, out, 0);
}